// HetersparseGAT_23862838297171
// MI455X (gfx1250) — compile-verified
//
#include <hip/hip_runtime.h>
#include <hip/hip_bf16.h>
#include <math.h>

typedef __attribute__((ext_vector_type(2))) float v2f;
typedef __attribute__((ext_vector_type(8))) float v8f;

namespace cfg {
constexpr int N       = 50000;
constexpr int N_USER  = 40000;
constexpr int E       = 400000;
constexpr int K       = 3;
constexpr int H       = 4;
constexpr int F_IN    = 128;
constexpr int F_OUT   = 64;
constexpr int NB      = 40;
constexpr float SLOPE = 0.2f;
}

// ---------------------------------------------------------------------------
// WMMA helper: D = A(16x4 f32) * B(4x16 f32) + C  via V_WMMA_F32_16X16X4_F32
// A lane layout: lane L holds row L%16, K = {0,1}+2*(L>=16)   (2 VGPRs)
// B lane layout: lane L holds col L%16, K = {0,1}+2*(L>=16)   (2 VGPRs)
// D lane layout: VGPR v -> row v + 8*(L>=16), col L%16        (8 VGPRs)
// ---------------------------------------------------------------------------
__device__ __forceinline__ v8f wmma4(v2f a, v2f b, v8f c) {
    return __builtin_amdgcn_wmma_f32_16x16x4_f32(
        false, a, false, b, (short)0, c, false, false);
}

// ---------------------------------------------------------------------------
// hp GEMM (specialized): one wave computes a 16-row strip x 128 columns
// (2 heads = 8 accumulators = 64 acc VGPRs -> no spills).  A (= h row strip)
// loaded once per k-step as a single b64 and reused by 8 WMMAs.
//   A = h [N x 128], B = w_k [H][128][64], C = hp [N][H*64]
// Wave id encodes (rowStrip, columnHalf).
// ---------------------------------------------------------------------------
__global__ void gat_hp_gemm(const float* __restrict__ A,
                            const float* __restrict__ B,
                            float* __restrict__ C) {
    using namespace cfg;
    const int wave = (int)((blockIdx.x * blockDim.x + threadIdx.x) >> 5);
    const int lane = threadIdx.x & 31;
    if (wave >= (N / 16) * 2) return;            // wave-uniform: EXEC stays all-ones

    const int strip    = wave >> 1;              // 16-row strip index
    const int headBase = (wave & 1) * 2;         // heads {0,1} or {2,3}

    const int r     = strip * 16 + (lane & 15);  // A row for this lane
    const int cl    = lane & 15;                 // column-in-tile for this lane
    const int khalf = (lane >> 4) << 1;          // 0 | 2

    v8f acc[8];
#pragma unroll
    for (int t = 0; t < 8; ++t) acc[t] = (v8f){};

    const v2f* Arow = (const v2f*)(A + (size_t)r * F_IN);
    const float* Bb = B + (size_t)headBase * (F_IN * F_OUT);
    for (int kk = 0; kk < F_IN; kk += 4) {
        const v2f a = Arow[(kk + khalf) >> 1];   // one b64 load, reused 8x
        v2f b[8];
#pragma unroll
        for (int t = 0; t < 8; ++t) {            // batched loads -> load clause
            const int head  = t >> 2;
            const int c     = ((t & 3) << 4) + cl;
            const size_t bb = (size_t)head * (F_IN * F_OUT)
                            + (size_t)(kk + khalf) * F_OUT + c;
            b[t][0] = Bb[bb];
            b[t][1] = Bb[bb + F_OUT];
        }
#pragma unroll
        for (int t = 0; t < 8; ++t)
            acc[t] = wmma4(a, b[t], acc[t]);
    }
    const int rbase = strip * 16 + ((lane >> 4) << 3);
#pragma unroll
    for (int t = 0; t < 8; ++t) {
        const int c = ((headBase + (t >> 2)) << 6) + ((t & 3) << 4) + cl;
#pragma unroll
        for (int v = 0; v < 8; ++v)
            C[(size_t)(rbase + v) * (H * F_OUT) + c] = acc[t][v];
    }
}

// ---------------------------------------------------------------------------
// Generic 16x64-per-wave WMMA GEMM: C[M x 64] = A[M x Kd] * B[Kd x 64]
// A loaded once per k-step (b64), reused by 4 WMMAs; B loads batched.
// ---------------------------------------------------------------------------
__global__ void gat_gemm16x64(const float* __restrict__ A,
                              const float* __restrict__ B,
                              float* __restrict__ C,
                              int M, int Kd, int lda, int ldb, int ldc) {
    const int wave = (int)((blockIdx.x * blockDim.x + threadIdx.x) >> 5);
    const int lane = threadIdx.x & 31;
    if (wave * 16 >= M) return;                  // wave-uniform

    const int r     = wave * 16 + (lane & 15);
    const int cl    = lane & 15;
    const int khalf = (lane >> 4) << 1;

    v8f acc[4];
#pragma unroll
    for (int t = 0; t < 4; ++t) acc[t] = (v8f){};

    const v2f* Arow = (const v2f*)(A + (size_t)r * lda);
    for (int kk = 0; kk < Kd; kk += 4) {
        const v2f a = Arow[(kk + khalf) >> 1];
        v2f b[4];
#pragma unroll
        for (int t = 0; t < 4; ++t) {
            const size_t bb = (size_t)(kk + khalf) * ldb + (t << 4) + cl;
            b[t][0] = B[bb];
            b[t][1] = B[bb + ldb];
        }
#pragma unroll
        for (int t = 0; t < 4; ++t)
            acc[t] = wmma4(a, b[t], acc[t]);
    }
    const int rbase = wave * 16 + ((lane >> 4) << 3);
#pragma unroll
    for (int t = 0; t < 4; ++t) {
        const int c = (t << 4) + cl;
#pragma unroll
        for (int v = 0; v < 8; ++v)
            C[(size_t)(rbase + v) * ldc + c] = acc[t][v];
    }
}

// ---------------------------------------------------------------------------
// Misc elementwise kernels
// ---------------------------------------------------------------------------
__global__ void gat_zero(float* __restrict__ p, size_t n) {
    size_t i = (size_t)blockIdx.x * blockDim.x + threadIdx.x;
    if (i < n) p[i] = 0.0f;
}
__global__ void gat_scale(float* __restrict__ p, float s, size_t n) {
    size_t i = (size_t)blockIdx.x * blockDim.x + threadIdx.x;
    if (i < n) p[i] *= s;
}

// s[n][h] = hp[n][h] . a_src[h] ; t likewise (per kind, a_* already k-offset)
__global__ void gat_st(const float* __restrict__ hp,
                       const float* __restrict__ a_src,
                       const float* __restrict__ a_trg,
                       float* __restrict__ s, float* __restrict__ t) {
    int i = blockIdx.x * blockDim.x + threadIdx.x;      // over N*H
    if (i >= cfg::N * cfg::H) return;
    int n = i >> 2, h = i & 3;
    const float* hpv = hp + (size_t)n * (cfg::H * cfg::F_OUT) + h * cfg::F_OUT;
    const float* as  = a_src + h * cfg::F_OUT;
    const float* at  = a_trg + h * cfg::F_OUT;
    float ss = 0.f, tt = 0.f;
#pragma unroll 8
    for (int f = 0; f < cfg::F_OUT; ++f) { float v = hpv[f]; ss += v * as[f]; tt += v * at[f]; }
    s[i] = ss; t[i] = tt;
}

__device__ __forceinline__ float gat_lrelu(float v) {
    return v >= 0.f ? v : cfg::SLOPE * v;
}

// Stage 1 global max of leaky-relu'd edge scores (grid-stride, block reduce)
__global__ void gat_emax_partial(const float* __restrict__ s, const float* __restrict__ t,
                                 const int* __restrict__ src, const int* __restrict__ trg,
                                 float* __restrict__ partials, int total) {
    __shared__ float red[256];
    float m = -3.4e38f;
    for (int i = blockIdx.x * blockDim.x + threadIdx.x; i < total;
         i += gridDim.x * blockDim.x) {
        int e = i >> 2, h = i & 3;
        float v = s[src[e] * 4 + h] + t[trg[e] * 4 + h];
        m = fmaxf(m, gat_lrelu(v));
    }
    red[threadIdx.x] = m; __syncthreads();
    for (int w = 128; w > 0; w >>= 1) {
        if (threadIdx.x < w) red[threadIdx.x] = fmaxf(red[threadIdx.x], red[threadIdx.x + w]);
        __syncthreads();
    }
    if (threadIdx.x == 0) partials[blockIdx.x] = red[0];
}
__global__ void gat_emax_final(const float* __restrict__ partials, float* __restrict__ emax, int n) {
    __shared__ float red[256];
    float m = -3.4e38f;
    for (int i = threadIdx.x; i < n; i += 256) m = fmaxf(m, partials[i]);
    red[threadIdx.x] = m; __syncthreads();
    for (int w = 128; w > 0; w >>= 1) {
        if (threadIdx.x < w) red[threadIdx.x] = fmaxf(red[threadIdx.x], red[threadIdx.x + w]);
        __syncthreads();
    }
    if (threadIdx.x == 0) emax[0] = red[0];
}

// denom[trg][h] += exp(lrelu(s[src]+t[trg]) - emax)
__global__ void gat_denom(const float* __restrict__ s, const float* __restrict__ t,
                          const int* __restrict__ src, const int* __restrict__ trg,
                          const float* __restrict__ emax, float* __restrict__ denom,
                          int total) {
    int i = blockIdx.x * blockDim.x + threadIdx.x;
    if (i >= total) return;
    int e = i >> 2, h = i & 3;
    int tn = trg[e];
    float v = s[src[e] * 4 + h] + t[tn * 4 + h];
    atomicAdd(&denom[tn * 4 + h], expf(gat_lrelu(v) - emax[0]));
}

// wave-per-edge: out[trg][f] += sum_h hp[src][h][f] * alpha[h]   (head mean folded later)
__global__ void gat_scatter(const float* __restrict__ hp,
                            const float* __restrict__ s, const float* __restrict__ t,
                            const int* __restrict__ src, const int* __restrict__ trg,
                            const float* __restrict__ emax, const float* __restrict__ denom,
                            float* __restrict__ outk) {
    int wave = (int)((blockIdx.x * blockDim.x + threadIdx.x) >> 5);
    int lane = threadIdx.x & 31;
    if (wave >= cfg::E) return;
    int sn = src[wave], tn = trg[wave];
    float em = emax[0];
    float alpha[cfg::H];
#pragma unroll
    for (int h = 0; h < cfg::H; ++h) {
        float v = s[sn * 4 + h] + t[tn * 4 + h];
        alpha[h] = expf(gat_lrelu(v) - em) / (denom[tn * 4 + h] + 1e-16f);
    }
    const float* hpn = hp + (size_t)sn * (cfg::H * cfg::F_OUT);
#pragma unroll
    for (int j = 0; j < 2; ++j) {
        int f = lane + j * 32;
        float acc = 0.f;
#pragma unroll
        for (int h = 0; h < cfg::H; ++h) acc += hpn[h * cfg::F_OUT + f] * alpha[h];
        atomicAdd(&outk[(size_t)tn * cfg::F_OUT + f], acc);
    }
}

// wave-per-node: q = tanh(fw1 + tw2), scores = q.m, beta = softmax_K, fusion = sum_k beta_k * ta_k
__global__ void gat_fusion(const float* __restrict__ ta,    // [K][N][64] (scaled out)
                           const float* __restrict__ tw2,   // [K][N][64]
                           const float* __restrict__ fw1,   // [N_USER][64]
                           const float* __restrict__ mvec,  // [64]
                           float* __restrict__ fus) {       // [N_USER][64]
    int wave = (int)((blockIdx.x * blockDim.x + threadIdx.x) >> 5);
    int lane = threadIdx.x & 31;
    if (wave >= cfg::N_USER) return;
    int n = wave;
    float f1a = fw1[(size_t)n * 64 + lane];
    float f1b = fw1[(size_t)n * 64 + lane + 32];
    float ma = mvec[lane], mb = mvec[lane + 32];
    float score[cfg::K];
#pragma unroll
    for (int k = 0; k < cfg::K; ++k) {
        size_t base = (size_t)k * cfg::N * 64 + (size_t)n * 64;
        float qa = tanhf(f1a + tw2[base + lane]);
        float qb = tanhf(f1b + tw2[base + lane + 32]);
        float p = qa * ma + qb * mb;
        for (int off = 16; off > 0; off >>= 1) p += __shfl_xor(p, off, 32);
        score[k] = p;
    }
    float mx = fmaxf(score[0], fmaxf(score[1], score[2]));
    float e0 = expf(score[0] - mx), e1 = expf(score[1] - mx), e2 = expf(score[2] - mx);
    float inv = 1.f / (e0 + e1 + e2);
    float b0 = e0 * inv, b1 = e1 * inv, b2 = e2 * inv;
#pragma unroll
    for (int j = 0; j < 2; ++j) {
        int f = lane + j * 32;
        size_t nb = (size_t)n * 64 + f;
        float v = b0 * ta[nb] + b1 * ta[(size_t)cfg::N * 64 + nb] + b2 * ta[(size_t)2 * cfg::N * 64 + nb];
        fus[nb] = v;
    }
}

// block-per-node: ret = concat(ta[0..2][n], fusion[n]) (256), logits = ret@fc_w+b, log_softmax
__global__ void gat_logits(const float* __restrict__ ta, const float* __restrict__ fus,
                           const float* __restrict__ fc_w, const float* __restrict__ fc_b,
                           float* __restrict__ outp) {
    __shared__ float ret[256];
    __shared__ float lg[cfg::NB];
    int n = blockIdx.x;
    int t = threadIdx.x;                 // blockDim == 256
    int k = t >> 6, f = t & 63;
    if (k < cfg::K) ret[t] = ta[(size_t)k * cfg::N * 64 + (size_t)n * 64 + f];
    else            ret[t] = fus[(size_t)n * 64 + f];
    __syncthreads();
    if (t < cfg::NB) {
        float acc = fc_b[t];
#pragma unroll 8
        for (int j = 0; j < 256; ++j) acc += ret[j] * fc_w[(size_t)j * cfg::NB + t];
        lg[t] = acc;
    }
    __syncthreads();
    if (t < cfg::NB) {
        float mx = -3.4e38f;
        for (int c = 0; c < cfg::NB; ++c) mx = fmaxf(mx, lg[c]);
        float sum = 0.f;
        for (int c = 0; c < cfg::NB; ++c) sum += expf(lg[c] - mx);
        outp[(size_t)n * cfg::NB + t] = lg[t] - mx - logf(sum);
    }
}

// ---------------------------------------------------------------------------
// Host-side launcher
// ---------------------------------------------------------------------------
extern "C" void kernel_launch(void* const* d_in, const int* in_sizes, int n_in,
                              void* d_out, int out_size, void* d_ws, size_t ws_size,
                              hipStream_t stream) {
    using namespace cfg;
    const float* h_ptr  = (const float*)d_in[0];
    const int*   eidx   = (const int*)  d_in[1];
    const float* w_ptr  = (const float*)d_in[2];
    const float* a_src  = (const float*)d_in[3];
    const float* a_trg  = (const float*)d_in[4];
    const float* w1     = (const float*)d_in[5];
    const float* w2     = (const float*)d_in[6];
    const float* mvec   = (const float*)d_in[7];
    const float* fc_w   = (const float*)d_in[8];
    const float* fc_b   = (const float*)d_in[9];
    float* outp = (float*)d_out;

    float* ws = (float*)d_ws;
    size_t off = 0;
    float* hp       = ws + off; off += (size_t)N * H * F_OUT;      // 12.8M
    float* sbuf     = ws + off; off += (size_t)N * H;              // 200k
    float* tbuf     = ws + off; off += (size_t)N * H;              // 200k
    float* denom    = ws + off; off += (size_t)N * H;              // 200k
    float* partials = ws + off; off += 1024;
    float* emax     = ws + off; off += 16;
    float* outb     = ws + off; off += (size_t)K * N * F_OUT;      // 9.6M
    float* tw2b     = ws + off; off += (size_t)K * N * F_OUT;      // 9.6M
    float* fw1b     = ws + off; off += (size_t)N_USER * F_OUT;     // 2.56M
    float* fusb     = ws + off; off += (size_t)N_USER * F_OUT;     // 2.56M

    auto nblk = [](size_t threads) { return (unsigned)((threads + 255) / 256); };

    // zero the cross-kind accumulator
    gat_zero<<<nblk((size_t)K * N * F_OUT), 256, 0, stream>>>(outb, (size_t)K * N * F_OUT);

    for (int k = 0; k < K; ++k) {
        // hp = h @ w[k] : one wave per (16-row strip, 2-head column half)
        gat_hp_gemm<<<nblk((size_t)(N / 16) * 2 * 32), 256, 0, stream>>>(
            h_ptr, w_ptr + (size_t)k * H * F_IN * F_OUT, hp);

        // attention coefficients
        gat_st<<<nblk((size_t)N * H), 256, 0, stream>>>(
            hp, a_src + (size_t)k * H * F_OUT, a_trg + (size_t)k * H * F_OUT, sbuf, tbuf);

        const int* src = eidx + (size_t)k * 2 * E;
        const int* trg = src + E;
        const int total = E * H;

        gat_emax_partial<<<1024, 256, 0, stream>>>(sbuf, tbuf, src, trg, partials, total);
        gat_emax_final<<<1, 256, 0, stream>>>(partials, emax, 1024);

        gat_zero<<<nblk((size_t)N * H), 256, 0, stream>>>(denom, (size_t)N * H);
        gat_denom<<<nblk((size_t)total), 256, 0, stream>>>(sbuf, tbuf, src, trg, emax, denom, total);

        gat_scatter<<<nblk((size_t)E * 32), 256, 0, stream>>>(
            hp, sbuf, tbuf, src, trg, emax, denom, outb + (size_t)k * N * F_OUT);
    }

    // head mean: out *= 1/H  ->  out == type_aware ([K][N][64], rows >= N_USER unused)
    gat_scale<<<nblk((size_t)K * N * F_OUT), 256, 0, stream>>>(outb, 1.0f / H, (size_t)K * N * F_OUT);

    // fw1 = h[:N_USER] @ w1
    gat_gemm16x64<<<nblk((size_t)(N_USER / 16) * 32), 256, 0, stream>>>(
        h_ptr, w1, fw1b, N_USER, F_IN, F_IN, F_OUT, F_OUT);

    // tw2 = type_aware @ w2   (flattened M = K*N rows of 64)
    gat_gemm16x64<<<nblk((size_t)(K * N / 16) * 32), 256, 0, stream>>>(
        outb, w2, tw2b, K * N, F_OUT, F_OUT, F_OUT, F_OUT);

    gat_fusion<<<nblk((size_t)N_USER * 32), 256, 0, stream>>>(outb, tw2b, fw1b, mvec, fusb);
    gat_logits<<<N_USER, 256, 0, stream>>>(outb, fusb, fc_w, fc_b, outp);

    (void)in_sizes; (void)n_in; (void)out_size; (void)ws_size;
}